// ModelNew_5909874999942
// MI455X (gfx1250) — compile-verified
//
#include <hip/hip_runtime.h>

typedef __attribute__((ext_vector_type(4))) float v4f;
typedef __attribute__((ext_vector_type(4))) int   v4i;
typedef __attribute__((address_space(1))) v4i g_v4i;   // global (device) memory
typedef __attribute__((address_space(3))) v4i l_v4i;   // LDS

// -------- capability probe: async global->LDS DMA (gfx1250 ASYNCcnt path) ----
#if __has_builtin(__builtin_amdgcn_global_load_async_to_lds_b128) && \
    __has_builtin(__builtin_amdgcn_s_wait_asynccnt)
#define HAVE_ASYNC_LDS 1
#else
#define HAVE_ASYNC_LDS 0
#endif

// Problem geometry (fixed by the reference): x is (B, N, C) f32, scan along N.
#define B 32
#define N 4096
#define C 512
#define S 16                 // segments per scan
#define L (N / S)            // 256 rows per segment
#define C4 (C / 4)           // 128 float4 columns per row
#define BLOCK 256
#define TOTAL_THREADS (B * S * C4)       // 65536 threads = 2048 wave32 waves
#define NBLOCKS (TOTAL_THREADS / BLOCK)  // 256 blocks
#define PF_DIST 24           // prefetch distance in rows (each row = fresh lines)

// ---------------------------------------------------------------------------
// Pass 1: thread owns (b, segment s, 4 channels). Computes 4 independent
// segment products over 256 rows. Wave footprint per step: 32 lanes x 16B =
// 512B contiguous -> perfectly coalesced b128 loads. Products -> d_ws (1 MB).
// ---------------------------------------------------------------------------
__global__ __launch_bounds__(BLOCK) void cumprod_segprod(const float* __restrict__ x,
                                                         float* __restrict__ ws) {
  unsigned tid = blockIdx.x * BLOCK + threadIdx.x;
  unsigned c4 = tid & (C4 - 1);
  unsigned s  = (tid >> 7) & (S - 1);
  unsigned b  = tid >> 11;

  const float* px = x + ((size_t)(b * N) + (size_t)s * L) * C + c4 * 4;

  v4f acc = {1.0f, 1.0f, 1.0f, 1.0f};
#pragma unroll 8
  for (int i = 0; i < L; ++i) {
    __builtin_prefetch(px + (size_t)(i + PF_DIST) * C, 0, 0);  // -> GL2
    acc *= *(const v4f*)(px + (size_t)i * C);
  }
  *(v4f*)(ws + (size_t)(b * S + s) * C + c4 * 4) = acc;
}

// ---------------------------------------------------------------------------
// Pass 2: block covers two consecutive segments (s_low, s_low+1) of one b,
// all 512 channels. The <=15 ws rows the block needs are DMA'd into LDS with
// global_load_async_to_lds_b128 (overlapped with prefetching the x stream),
// then each thread builds its exclusive prefix from LDS and re-scans its
// segment, emitting the inclusive cumprod through non-temporal b128 stores.
// Blocks run in REVERSE order so pass 1's L2-resident tail of x is read first.
// ---------------------------------------------------------------------------
__global__ __launch_bounds__(BLOCK) void cumprod_apply(const float* __restrict__ x,
                                                       const float* __restrict__ ws,
                                                       float* __restrict__ out) {
  __shared__ v4f lds_w[(S - 1) * C4];   // up to 15 staged ws rows = 30 KB

  unsigned bid  = (unsigned)(gridDim.x - 1u - blockIdx.x);  // reverse block order
  unsigned ltid = threadIdx.x;
  unsigned tid  = bid * BLOCK + ltid;
  unsigned c4 = tid & (C4 - 1);
  unsigned s  = (tid >> 7) & (S - 1);
  unsigned b  = (bid * 2) >> 4;         // shared by whole block
  unsigned s_low = (bid * 2) & (S - 1); // lower segment in this block
  unsigned nq = (s_low + 1) * C4;       // float4 count of ws rows 0..s_low

  const v4f* wsrc = (const v4f*)(ws + (size_t)b * S * C);

  // ---- stage needed ws rows into LDS (async DMA, ASYNCcnt-tracked) ----
#if HAVE_ASYNC_LDS
  for (unsigned idx = ltid; idx < nq; idx += BLOCK) {
    __builtin_amdgcn_global_load_async_to_lds_b128(
        (g_v4i*)(void*)(wsrc + idx),
        (l_v4i*)(void*)(lds_w + idx),
        /*offset=*/0, /*cpol=*/0);
  }
#else
  for (unsigned idx = ltid; idx < nq; idx += BLOCK) lds_w[idx] = wsrc[idx];
#endif

  // overlap: warm the x stream for this thread's segment while the DMA runs
  const float* px = x + ((size_t)(b * N) + (size_t)s * L) * C + c4 * 4;
  float* po       = out + ((size_t)(b * N) + (size_t)s * L) * C + c4 * 4;
#pragma unroll
  for (int i = 0; i < PF_DIST; ++i)
    __builtin_prefetch(px + (size_t)i * C, 0, 0);

#if HAVE_ASYNC_LDS
  __builtin_amdgcn_s_wait_asynccnt(0);  // this wave's DMA writes landed in LDS
#endif
  __syncthreads();                      // cross-wave visibility

  // ---- exclusive prefix over segment products (ds_load_b128, conflict-free) ----
  v4f acc = {1.0f, 1.0f, 1.0f, 1.0f};
  for (unsigned t = 0; t < s; ++t) acc *= lds_w[t * C4 + c4];

  // ---- re-scan the segment; NT stores keep x resident in L2 ----
#pragma unroll 8
  for (int i = 0; i < L; ++i) {
    __builtin_prefetch(px + (size_t)(i + PF_DIST) * C, 0, 0);
    acc *= *(const v4f*)(px + (size_t)i * C);
    __builtin_nontemporal_store(acc, (v4f*)(po + (size_t)i * C));
  }
}

// ---------------------------------------------------------------------------
// Harness entry point. d_in[0] = x (f32), d_in[1] = dim (==1, ignored).
// d_ws needs B*S*C*4 = 1 MB of scratch for segment products.
// ---------------------------------------------------------------------------
extern "C" void kernel_launch(void* const* d_in, const int* in_sizes, int n_in,
                              void* d_out, int out_size, void* d_ws, size_t ws_size,
                              hipStream_t stream) {
  const float* x = (const float*)d_in[0];
  float* out = (float*)d_out;
  float* ws = (float*)d_ws;   // B*S*C floats = 1 MB

  (void)in_sizes; (void)n_in; (void)out_size; (void)ws_size;

  hipLaunchKernelGGL(cumprod_segprod, dim3(NBLOCKS), dim3(BLOCK), 0, stream, x, ws);
  hipLaunchKernelGGL(cumprod_apply,   dim3(NBLOCKS), dim3(BLOCK), 0, stream, x, ws, out);
}